// RGNNLayer_14070312862199
// MI455X (gfx1250) — compile-verified
//
#include <hip/hip_runtime.h>
#include <hip/hip_bf16.h>

// ---------------------------------------------------------------------------
// RGNN layer for MI455X (gfx1250):
//   out = x @ W_root + b_root + sum_l scatter_add(x @ W_convs[l], edges_l)
//
// Memory-bound (~1 GB moved vs 65 GFLOP) -> full f32 via V_WMMA_F32_16X16X4_F32
// (exact reference precision, still on the matrix pipe). W is staged through
// LDS with the gfx1250 async Global->LDS path (ASYNCcnt) and double-buffered
// so the next K-chunk streams in while the current one feeds the WMMAs.
// ---------------------------------------------------------------------------

typedef __attribute__((ext_vector_type(2))) float v2f;
typedef __attribute__((ext_vector_type(8))) float v8f;

#define DIM 256
#define ROWS_PER_BLOCK 128   // 8 row-strips of 16
#define THREADS_GEMM 512     // 16 waves: wave = (row_strip, n_half)
#define KCHUNK 64
// LDS row stride (floats). 248 is chosen so that:
//  * ds_load_2addr_b32 can fetch {row k, row k+1} in ONE op: offset1 = 248 <= 255
//    -> B-frag lands in a consecutive VGPR pair, no repack movs, per-op dscnt waits
//  * 2*248 mod 64 = 48 in [16,48] -> the two half-waves' 16-bank groups are disjoint
//  * 248*4 = 992 B is 16B-aligned -> b128 staging stays aligned
#define LDS_STRIDE 248
#define CHUNK_FLOATS (KCHUNK * LDS_STRIDE)      // 15,872 floats = 63,488 B
#define GEMM_SMEM (2 * CHUNK_FLOATS * 4)        // 126,976 B dynamic LDS (WGP has 320 KB)

#if __has_builtin(__builtin_amdgcn_global_load_async_to_lds_b128)
#define HAVE_ASYNC_LDS 1
// The builtin's params are generic (vector_size) int4 pointers in as1/as3:
//   (v4i addrspace(1)*, v4i addrspace(3)*, imm int offset, imm int cpol)
typedef int async_b128_t __attribute__((vector_size(16)));
typedef __attribute__((address_space(1))) async_b128_t* gptr_b128;
typedef __attribute__((address_space(3))) async_b128_t* lptr_b128;
#else
#define HAVE_ASYNC_LDS 0
#endif

// Double-buffered W chunks: buf[b][r * LDS_STRIDE + c] = W[kb + r, c]
extern __shared__ float lds_w[];

// Issue the copy of W[kb .. kb+KCHUNK) into an LDS chunk buffer.
// 64 rows * 64 float4 = 4096 xfers / 512 threads = 8 per thread.
__device__ __forceinline__ void stage_chunk(float* __restrict__ buf,
                                            const float* __restrict__ W,
                                            int kb, int tid) {
    #pragma unroll
    for (int i = 0; i < 8; ++i) {
        const int idx = tid + i * THREADS_GEMM;
        const int r   = idx >> 6;
        const int c   = (idx & 63) << 2;
        const float* src = W + (size_t)(kb + r) * DIM + c;
        float*       dst = &buf[r * LDS_STRIDE + c];
#if HAVE_ASYNC_LDS
        // global_load_async_to_lds_b128: LDS written directly, tracked by ASYNCcnt
        async_b128_t* s_gen = (async_b128_t*)const_cast<float*>(src);
        async_b128_t* d_gen = (async_b128_t*)dst;
        __builtin_amdgcn_global_load_async_to_lds_b128(
            (gptr_b128)s_gen, (lptr_b128)d_gen, 0, 0);
#else
        *(float4*)dst = *(const float4*)src;
#endif
    }
}

__device__ __forceinline__ void wait_stage() {
#if HAVE_ASYNC_LDS
# if __has_builtin(__builtin_amdgcn_s_wait_asynccnt)
    __builtin_amdgcn_s_wait_asynccnt(0);
# else
    asm volatile("s_wait_asynccnt 0x0" ::: "memory");
# endif
#endif
}

__global__ void __launch_bounds__(THREADS_GEMM)
rgnn_gemm_wmma(const float* __restrict__ X,     // [nrows, 256] row major
               const float* __restrict__ W,     // [256, 256]  row major (x @ W)
               const float* __restrict__ bias,  // [256] or nullptr
               float* __restrict__ Y,           // [nrows, 256]
               int nrows) {
    const int lane = threadIdx.x & 31;
    const int wave = threadIdx.x >> 5;
    const int lhi  = lane >> 4;      // half-wave id (0/1)
    const int llo  = lane & 15;      // position within half-wave

    const int block_row = blockIdx.x * ROWS_PER_BLOCK;
    const int m0     = block_row + (wave >> 1) * 16;   // 16-row strip owned by wave
    const int n_half = (wave & 1) * 128;               // 8 N-tiles of 16

    // A row pointer for this lane (clamped for the ragged tail; stores are guarded)
    int arow_idx = m0 + llo;
    if (arow_idx > nrows - 1) arow_idx = nrows - 1;
    const float* __restrict__ arow = X + (size_t)arow_idx * DIM;

    v8f acc[8];
    #pragma unroll
    for (int t = 0; t < 8; ++t) acc[t] = (v8f)0.0f;

    float* const buf0 = lds_w;
    float* const buf1 = lds_w + CHUNK_FLOATS;

    // Prologue: stream in chunk 0.
    stage_chunk(buf0, W, 0, threadIdx.x);
    wait_stage();
    __syncthreads();

    int pb = 0;
    for (int kb = 0; kb < DIM; kb += KCHUNK, pb ^= 1) {
        // Kick off the next chunk into the other buffer before computing.
        if (kb + KCHUNK < DIM)
            stage_chunk(pb ? buf0 : buf1, W, kb + KCHUNK, threadIdx.x);

        const float* __restrict__ cw = pb ? buf1 : buf0;
        for (int k = 0; k < KCHUNK; k += 4) {
            // A frag (16x4 f32 layout): lanes 0-15 carry K=k,k+1 ; lanes 16-31 K=k+2,k+3
            const v2f a = *(const v2f*)(arow + kb + k + 2 * lhi);
            const float* __restrict__ brow = &cw[(k + 2 * lhi) * LDS_STRIDE];
            #pragma unroll
            for (int t = 0; t < 8; ++t) {
                const int col = n_half + t * 16 + llo;
                v2f b;
                b.x = brow[col];               // row K=k   (+2*lhi)
                b.y = brow[col + LDS_STRIDE];  // row K=k+1 (+2*lhi)  (one ds_load_2addr)
                acc[t] = __builtin_amdgcn_wmma_f32_16x16x4_f32(
                    false, a, false, b, (short)0, acc[t], false, false);
            }
        }

        // Next chunk's async writes complete + all waves done reading this buffer.
        wait_stage();
        __syncthreads();
    }

    // Bias + store. D layout: VGPR v, lanes 0-15 -> (M=m0+v, N=n0+llo),
    // lanes 16-31 -> (M=m0+8+v, N=n0+llo).
    #pragma unroll
    for (int t = 0; t < 8; ++t) {
        const int col = n_half + t * 16 + llo;
        const float bb = bias ? bias[col] : 0.0f;
        #pragma unroll
        for (int v = 0; v < 8; ++v) {
            const int row = m0 + v + 8 * lhi;
            if (row < nrows) Y[(size_t)row * DIM + col] = acc[t][v] + bb;
        }
    }
}

// ---------------------------------------------------------------------------
// Edge scatter: one wave per edge; lane handles 8 consecutive floats
// (2x float4 loads of h[src], 8x global_atomic_add_f32 into out[dst]).
// h and out both fit in the 192 MB L2, so atomics resolve at L2.
// ---------------------------------------------------------------------------
__device__ __forceinline__ void atom_add_f32(float* p, float v) {
    unsafeAtomicAdd(p, v);   // hardware global_atomic_add_f32
}

__global__ void rgnn_edge_scatter(const int* __restrict__ src,
                                  const int* __restrict__ dst,
                                  const float* __restrict__ h,   // [nrows, 256]
                                  float* __restrict__ out,       // [nrows, 256]
                                  int nedges) {
    const int lane   = threadIdx.x & 31;
    const int wave   = (blockIdx.x * blockDim.x + threadIdx.x) >> 5;
    const int nwaves = (gridDim.x * blockDim.x) >> 5;

    for (int e = wave; e < nedges; e += nwaves) {
        const int s = src[e];
        const int d = dst[e];
        const float4* __restrict__ hp = (const float4*)(h + (size_t)s * DIM);
        float* __restrict__ op = out + (size_t)d * DIM + lane * 8;
        const float4 v0 = hp[lane * 2 + 0];   // cols lane*8 .. lane*8+3
        const float4 v1 = hp[lane * 2 + 1];   // cols lane*8+4 .. lane*8+7
        atom_add_f32(op + 0, v0.x);
        atom_add_f32(op + 1, v0.y);
        atom_add_f32(op + 2, v0.z);
        atom_add_f32(op + 3, v0.w);
        atom_add_f32(op + 4, v1.x);
        atom_add_f32(op + 5, v1.y);
        atom_add_f32(op + 6, v1.z);
        atom_add_f32(op + 7, v1.w);
    }
}

// ---------------------------------------------------------------------------
// Launch: root GEMM (fused bias) -> per label {conv GEMM into ws, scatter}.
// Stream order provides all dependencies; ws holds one h buffer (102.4 MB).
// ---------------------------------------------------------------------------
extern "C" void kernel_launch(void* const* d_in, const int* in_sizes, int n_in,
                              void* d_out, int out_size, void* d_ws, size_t ws_size,
                              hipStream_t stream) {
    const float* x  = (const float*)d_in[0];   // [nrows, 256]
    const int*   ei = (const int*)d_in[1];     // [4, 2, E]
    const float* Wr = (const float*)d_in[2];   // [256, 256]
    const float* br = (const float*)d_in[3];   // [256]
    const float* Wc = (const float*)d_in[4];   // [4, 256, 256]
    float* out = (float*)d_out;
    float* h   = (float*)d_ws;                 // one [nrows, 256] f32 buffer

    const int nrows = in_sizes[0] / DIM;
    const int E     = in_sizes[1] / (4 * 2);
    const int gemm_blocks = (nrows + ROWS_PER_BLOCK - 1) / ROWS_PER_BLOCK;

    // Allow 124 KB of dynamic LDS (deterministic host-side call, capture-safe).
    (void)hipFuncSetAttribute((const void*)rgnn_gemm_wmma,
                              hipFuncAttributeMaxDynamicSharedMemorySize, GEMM_SMEM);

    // out = x @ W_root + b_root
    rgnn_gemm_wmma<<<gemm_blocks, THREADS_GEMM, GEMM_SMEM, stream>>>(x, Wr, br, out, nrows);

    for (int l = 0; l < 4; ++l) {
        // h = x @ W_convs[l]
        rgnn_gemm_wmma<<<gemm_blocks, THREADS_GEMM, GEMM_SMEM, stream>>>(
            x, Wc + (size_t)l * DIM * DIM, nullptr, h, nrows);
        // out[dst] += h[src]
        const int* src = ei + (size_t)l * 2 * E;
        const int* dst = src + E;
        rgnn_edge_scatter<<<2048, 256, 0, stream>>>(src, dst, h, out, E);
    }
}